// SceneFlowLossModel_23304492548685
// MI455X (gfx1250) — compile-verified
//
#include <hip/hip_runtime.h>
#include <hip/hip_bf16.h>
#include <math.h>

typedef __attribute__((ext_vector_type(2))) float v2f;
typedef __attribute__((ext_vector_type(8))) float v8f;

#define EPS_W 1e-8f

// store->load fence within a wave: conservative hardware wait (LLVM uses the
// same lowering for intra-wave __syncthreads) plus scheduling barriers.
__device__ __forceinline__ void wave_store_load_fence() {
    __builtin_amdgcn_wave_barrier();
    asm volatile("s_wait_dscnt 0x0" ::: "memory");
    __builtin_amdgcn_wave_barrier();
}
// load->store (WAR) fence: same-wave LDS ops execute in order (ISA 7.3),
// so only the compiler must be prevented from reordering.
__device__ __forceinline__ void compiler_lds_fence() {
    __builtin_amdgcn_wave_barrier();
    asm volatile("" ::: "memory");
    __builtin_amdgcn_wave_barrier();
}

template<int K>
__device__ __forceinline__ void topk_insert(float (&bd)[K], int (&bi)[K], float d, int j) {
    if (d < bd[K - 1]) {
        bd[K - 1] = d; bi[K - 1] = j;
#pragma unroll
        for (int t = K - 1; t > 0; --t) {
            if (bd[t] < bd[t - 1]) {
                float td = bd[t]; bd[t] = bd[t - 1]; bd[t - 1] = td;
                int   ti = bi[t]; bi[t] = bi[t - 1]; bi[t - 1] = ti;
            }
        }
    }
}

// ---------------------------------------------------------------------------
// WMMA-tiled k-NN. 256 threads = 8 waves; each wave owns one 16-query row
// tile; the block shares the key set staged in LDS (<=48KB of 320KB WGP LDS).
// d(m,n) = aa_m + bb_n - 2*dot via V_WMMA_F32_16X16X4_F32 per 16x16 tile:
//   A row m = (-2x,-2y,-2z,aa_m), B col n = (x',y',z',1) -> D = aa - 2*dot.
// Two subtiles (32 keys) per iteration; both half-waves scan 8 columns each
// via float4 LDS reads; per-row half-lists merged once at the end.
// ---------------------------------------------------------------------------
template<int K, bool WRITE_IDX, bool WRITE_DIST, bool ACCUM_MIN>
__global__ __launch_bounds__(256)
void knn_wmma_kernel(const float* __restrict__ Qp, const float* __restrict__ Qa,
                     const float* __restrict__ Kp, const float* __restrict__ Ka,
                     int N, int M,
                     int* __restrict__ idx_out, float* __restrict__ dist_out,
                     float* __restrict__ accum, float accum_scale)
{
    extern __shared__ float smem[];
    float* keys = smem;                               // M*3 staged key coords
    const int tid  = threadIdx.x;
    const int lane = tid & 31;
    const int wave = tid >> 5;
    const int m    = lane & 15;
    const int hi   = lane >> 4;
    const int b    = blockIdx.y;
    float* scratch = smem + M * 3 + wave * 640;       // 2 tile slots (stride-20 rows)

    // ---- stage key set into LDS (shared by all 8 waves) --------------------
    if (Ka == nullptr) {
        // CDNA5 async copy global->LDS: 16B/lane, ASYNCcnt tracked, no VGPRs.
        const float* kbase = Kp + (size_t)b * M * 3;
        const uint32_t lds0 = (uint32_t)(uintptr_t)keys;  // flat LDS addr [31:0] == LDS offset
        const int nbytes = M * 12;
        for (int off = tid * 16; off < nbytes; off += 256 * 16) {
            uint32_t dst = lds0 + (uint32_t)off;
            asm volatile("global_load_async_to_lds_b128 %0, %1, %2"
                         :: "v"(dst), "v"(off), "s"(kbase) : "memory");
        }
        asm volatile("s_wait_asynccnt 0x0" ::: "memory");
    } else {
        // displaced keys (warp = pc1 + flow) need VALU: stage via registers
        for (int i = tid; i < M; i += 256) {
            const int gi = (b * M + i) * 3;
            keys[i * 3 + 0] = Kp[gi + 0] + Ka[gi + 0];
            keys[i * 3 + 1] = Kp[gi + 1] + Ka[gi + 1];
            keys[i * 3 + 2] = Kp[gi + 2] + Ka[gi + 2];
        }
    }
    __syncthreads();

    // ---- this wave's 16 query points (lane role m owns query row m) --------
    const int i0 = (blockIdx.x * 8 + wave) * 16;
    const int qi = (b * N + i0 + m) * 3;
    float qx = Qp[qi + 0], qy = Qp[qi + 1], qz = Qp[qi + 2];
    if (Qa) { qx += Qa[qi + 0]; qy += Qa[qi + 1]; qz += Qa[qi + 2]; }
    const float aa = qx * qx + qy * qy + qz * qz;

    // A 16x4 f32 layout: lanes 0-15 -> (K0,K1), lanes 16-31 -> (K2,K3)
    v2f avec;
    if (hi == 0) { avec[0] = -2.0f * qx; avec[1] = -2.0f * qy; }
    else         { avec[0] = -2.0f * qz; avec[1] = aa;         }

    float bestd[K];
    int   besti[K];
#pragma unroll
    for (int t = 0; t < K; ++t) { bestd[t] = 3.4e38f; besti[t] = 0; }

    const int rowbase = m * 20 + hi * 8;   // this lane's scan base within a tile slot

    for (int j0 = 0; j0 < M; j0 += 32) {
        // ---- compute two 16x16 distance subtiles -----------------------
#pragma unroll
        for (int s = 0; s < 2; ++s) {
            const int jj = j0 + s * 16;
            const float kx = keys[(jj + m) * 3 + 0];
            const float ky = keys[(jj + m) * 3 + 1];
            const float kz = keys[(jj + m) * 3 + 2];
            const float bb = kx * kx + ky * ky + kz * kz;

            // B 4x16 f32: lanes 0-15 -> rows (K0,K1), lanes 16-31 -> (K2,K3)
            v2f bvec;
            if (hi == 0) { bvec[0] = kx; bvec[1] = ky;   }
            else         { bvec[0] = kz; bvec[1] = 1.0f; }

            v8f c = {};
            // uniform control flow: EXEC all-1s as WMMA requires
            c = __builtin_amdgcn_wmma_f32_16x16x4_f32(
                    false, avec, false, bvec, (short)0, c, false, false);

            // C/D layout: VGPR v -> row v + 8*hi, column = lane&15
            float* slot = scratch + s * 320;
#pragma unroll
            for (int v = 0; v < 8; ++v)
                slot[(v + 8 * hi) * 20 + m] = c[v] + bb;
        }
        wave_store_load_fence();

        // ---- scan: lane owns row m, columns [hi*8, hi*8+8) of each subtile
#pragma unroll
        for (int s = 0; s < 2; ++s) {
            const float* slot = scratch + s * 320;
            const float4 d0 = *reinterpret_cast<const float4*>(&slot[rowbase]);
            const float4 d1 = *reinterpret_cast<const float4*>(&slot[rowbase + 4]);
            const int jb = j0 + s * 16 + hi * 8;
            topk_insert<K>(bestd, besti, d0.x, jb + 0);
            topk_insert<K>(bestd, besti, d0.y, jb + 1);
            topk_insert<K>(bestd, besti, d0.z, jb + 2);
            topk_insert<K>(bestd, besti, d0.w, jb + 3);
            topk_insert<K>(bestd, besti, d1.x, jb + 4);
            topk_insert<K>(bestd, besti, d1.y, jb + 5);
            topk_insert<K>(bestd, besti, d1.z, jb + 6);
            topk_insert<K>(bestd, besti, d1.w, jb + 7);
        }
        compiler_lds_fence();   // WAR only: same-wave LDS ops are in-order
    }

    // ---- merge the two half-lists per row ----------------------------------
    float* mD = scratch;                 // 16*K floats
    int*   mI = (int*)(scratch + 160);   // 16*K ints
    if (hi == 1) {
#pragma unroll
        for (int t = 0; t < K; ++t) { mD[m * K + t] = bestd[t]; mI[m * K + t] = besti[t]; }
    }
    wave_store_load_fence();

    if (hi == 0) {
#pragma unroll
        for (int t = 0; t < K; ++t)     // insert partner's sorted list
            topk_insert<K>(bestd, besti, mD[m * K + t], mI[m * K + t]);

        const int row = b * N + i0 + m;
        if (WRITE_IDX) {
#pragma unroll
            for (int t = 0; t < K; ++t) idx_out[row * K + t] = besti[t];
        }
        if (WRITE_DIST) {
#pragma unroll
            for (int t = 0; t < K; ++t) dist_out[row * K + t] = bestd[t];
        }
        if (ACCUM_MIN) atomicAdd(accum, accum_scale * bestd[0]);
    }
}

// ---------------------------------------------------------------------------
// curvature of pc2: curv[n] = (sum_{t<10} pc2[idx[t]] - 10*pc2[n]) / 9
// ---------------------------------------------------------------------------
__global__ void curv_kernel(const float* __restrict__ pts,
                            const int* __restrict__ idx,
                            float* __restrict__ curv, int N, int P)
{
    int tid = blockIdx.x * blockDim.x + threadIdx.x;
    if (tid >= P) return;
    const int b = tid / N;
    const float* base = pts + (size_t)b * N * 3;
    const float px = pts[tid * 3 + 0], py = pts[tid * 3 + 1], pz = pts[tid * 3 + 2];
    float sx = 0.f, sy = 0.f, sz = 0.f;
#pragma unroll
    for (int t = 0; t < 10; ++t) {
        const int j = idx[tid * 10 + t];
        sx += base[j * 3 + 0]; sy += base[j * 3 + 1]; sz += base[j * 3 + 2];
    }
    const float inv9 = 1.0f / 9.0f;
    curv[tid * 3 + 0] = (sx - 10.f * px) * inv9;
    curv[tid * 3 + 1] = (sy - 10.f * py) * inv9;
    curv[tid * 3 + 2] = (sz - 10.f * pz) * inv9;
}

// ---------------------------------------------------------------------------
// moved curvature (gather warp at pc1-self knn) + smoothness (k=9 prefix)
// ---------------------------------------------------------------------------
__global__ void moved_smooth_kernel(const float* __restrict__ pc1,
                                    const float* __restrict__ flow,
                                    const int* __restrict__ idx1,
                                    float* __restrict__ moved,
                                    float* __restrict__ out, float s_sm,
                                    int N, int P)
{
    int tid = blockIdx.x * blockDim.x + threadIdx.x;
    if (tid >= P) return;
    const int b = tid / N;
    const float* bp = pc1  + (size_t)b * N * 3;
    const float* bf = flow + (size_t)b * N * 3;
    const float wx = pc1[tid * 3 + 0] + flow[tid * 3 + 0];
    const float wy = pc1[tid * 3 + 1] + flow[tid * 3 + 1];
    const float wz = pc1[tid * 3 + 2] + flow[tid * 3 + 2];
    const float fx = flow[tid * 3 + 0], fy = flow[tid * 3 + 1], fz = flow[tid * 3 + 2];
    float sx = 0.f, sy = 0.f, sz = 0.f, sm = 0.f;
#pragma unroll
    for (int t = 0; t < 10; ++t) {
        const int j = idx1[tid * 10 + t];
        const float gx = bp[j * 3 + 0] + bf[j * 3 + 0];
        const float gy = bp[j * 3 + 1] + bf[j * 3 + 1];
        const float gz = bp[j * 3 + 2] + bf[j * 3 + 2];
        sx += gx - wx; sy += gy - wy; sz += gz - wz;
        if (t < 9) {  // smoothness uses the k=9 prefix (same ordering)
            const float dx = bf[j * 3 + 0] - fx;
            const float dy = bf[j * 3 + 1] - fy;
            const float dz = bf[j * 3 + 2] - fz;
            const float ss = dx * dx + dy * dy + dz * dz;
            sm += (ss > 0.f) ? sqrtf(ss) : 0.f;   // safe_norm
        }
    }
    const float inv9 = 1.0f / 9.0f;
    moved[tid * 3 + 0] = sx * inv9;
    moved[tid * 3 + 1] = sy * inv9;
    moved[tid * 3 + 2] = sz * inv9;
    atomicAdd(out, s_sm * (sm * 0.125f));
}

// ---------------------------------------------------------------------------
// interpolate curv2 at warp points (k=5 IDW), accumulate curvature + chamfer d1
// ---------------------------------------------------------------------------
__global__ void inter_kernel(const float* __restrict__ curv2,
                             const float* __restrict__ moved,
                             const float* __restrict__ dist12,
                             const int* __restrict__ idx12,
                             float* __restrict__ out,
                             float s_cu, float s_ch, int N, int P)
{
    int tid = blockIdx.x * blockDim.x + threadIdx.x;
    if (tid >= P) return;
    const int b = tid / N;
    const float* cb = curv2 + (size_t)b * N * 3;
    float w[5], wsum = 0.f;
#pragma unroll
    for (int t = 0; t < 5; ++t) { w[t] = 1.0f / (dist12[tid * 5 + t] + EPS_W); wsum += w[t]; }
    const float inv = 1.0f / wsum;
    float ix = 0.f, iy = 0.f, iz = 0.f;
#pragma unroll
    for (int t = 0; t < 5; ++t) {
        const int j = idx12[tid * 5 + t];
        const float ww = w[t] * inv;
        ix += ww * cb[j * 3 + 0];
        iy += ww * cb[j * 3 + 1];
        iz += ww * cb[j * 3 + 2];
    }
    const float dx = ix - moved[tid * 3 + 0];
    const float dy = iy - moved[tid * 3 + 1];
    const float dz = iz - moved[tid * 3 + 2];
    atomicAdd(out, s_cu * (dx * dx + dy * dy + dz * dz) + s_ch * dist12[tid * 5 + 0]);
}

__global__ void init_kernel(float* out) { out[0] = 0.0f; }

// ---------------------------------------------------------------------------
extern "C" void kernel_launch(void* const* d_in, const int* in_sizes, int n_in,
                              void* d_out, int out_size, void* d_ws, size_t ws_size,
                              hipStream_t stream)
{
    (void)n_in; (void)out_size; (void)ws_size;
    const int Ns[4]   = {4096, 2048, 1024, 512};
    const float wt[4] = {0.02f, 0.04f, 0.08f, 0.16f};
    const int B = in_sizes[0] / (3 * Ns[0]);
    float* out = (float*)d_out;

    init_kernel<<<1, 1, 0, stream>>>(out);

    for (int l = 0; l < 4; ++l) {
        const int N = Ns[l];
        const int P = B * N;
        const float* pc1  = (const float*)d_in[l];
        const float* pc2  = (const float*)d_in[4 + l];
        const float* flow = (const float*)d_in[8 + l];

        // workspace carve-up (reused each level; stream-serialized)
        int*   idx1   = (int*)d_ws;               // 10*P
        int*   idx2   = idx1 + 10 * P;            // 10*P
        int*   idx12  = idx2 + 10 * P;            // 5*P
        float* dist12 = (float*)(idx12 + 5 * P);  // 5*P
        float* curv2  = dist12 + 5 * P;           // 3*P
        float* moved  = curv2 + 3 * P;            // 3*P

        const float s_sm = wt[l] / (float)B;          // f_smoothness = 1.0
        const float s_ch = wt[l] / (float)B;          // f_chamfer    = 1.0
        const float s_cu = 0.3f * wt[l] / (float)B;   // f_curvature  = 0.3

        dim3 grid(N / 128, B), blk(256);
        const size_t shmem = (size_t)N * 12 + 8 * 640 * 4;  // keys + per-wave scratch

        knn_wmma_kernel<10, true, false, false><<<grid, blk, shmem, stream>>>(
            pc1, nullptr, pc1, nullptr, N, N, idx1, nullptr, nullptr, 0.f);
        knn_wmma_kernel<10, true, false, false><<<grid, blk, shmem, stream>>>(
            pc2, nullptr, pc2, nullptr, N, N, idx2, nullptr, nullptr, 0.f);
        knn_wmma_kernel<5, true, true, false><<<grid, blk, shmem, stream>>>(
            pc1, flow, pc2, nullptr, N, N, idx12, dist12, nullptr, 0.f);
        knn_wmma_kernel<1, false, false, true><<<grid, blk, shmem, stream>>>(
            pc2, nullptr, pc1, flow, N, N, nullptr, nullptr, out, s_ch);

        const int pb = 256, pg = (P + pb - 1) / pb;
        curv_kernel<<<pg, pb, 0, stream>>>(pc2, idx2, curv2, N, P);
        moved_smooth_kernel<<<pg, pb, 0, stream>>>(pc1, flow, idx1, moved, out, s_sm, N, P);
        inter_kernel<<<pg, pb, 0, stream>>>(curv2, moved, dist12, idx12, out, s_cu, s_ch, N, P);
    }
}